// RelativeLabelLossV2_14319420965547
// MI455X (gfx1250) — compile-verified
//
#include <hip/hip_runtime.h>
#include <hip/hip_bf16.h>
#include <math.h>

// ---------------------------------------------------------------------------
// RelativeLabelLoss on MI455X (gfx1250).
// B=2048 rows, C=32000 cols, f32. Memory-bound: 262 MB streamed once
// (~11 us at 23.3 TB/s). One 256-thread block (8 wave32) per row.
// Streaming uses the CDNA5 async global->LDS engine (ASYNCcnt), 2-stage
// software pipeline, each thread consumes exactly the LDS it filled
// (no barriers in the stream loop; s_wait_asynccnt sequences stages).
// NT cache policy: the 262 MB stream has zero reuse and exceeds the
// 192 MB L2, so loads carry TH=NT.
// Masked logsumexp = full online softmax minus the <=8 deduplicated
// excluded terms (identical to the reference's -BIG masking).
// ---------------------------------------------------------------------------

#define B_ROWS   2048
#define C_COLS   32000
#define GAMMA_   0.2f
#define BIG_     100000000.0f

#define TPB      256
#define CHUNK    4096          // floats per pipeline stage (16 KB)
#define NSTAGES  8             // ceil(32000 / 4096)

// gfx12 CPol: TH in [2:0]; TH_NT = 1 (non-temporal: single-use stream).
#define STREAM_CPOL 1

#if defined(__has_builtin)
#if __has_builtin(__builtin_amdgcn_global_load_async_to_lds_b128)
#define ASYNC_LDS 1
#endif
#endif
#ifndef ASYNC_LDS
#define ASYNC_LDS 0
#endif

// The b128 async-load builtin is typed on 4-int vectors (per hipcc diagnostic):
//   void(v4i addrspace(1)* src, v4i addrspace(3)* dst, imm int off, imm int cpol)
typedef int v4i __attribute__((vector_size(16)));
typedef __attribute__((address_space(1))) v4i* gv4i_p;   // global v4i*
typedef __attribute__((address_space(3))) v4i* lv4i_p;   // LDS v4i*

template <int N>
__device__ __forceinline__ void wait_async() {
#if ASYNC_LDS
#if __has_builtin(__builtin_amdgcn_s_wait_asynccnt)
  __builtin_amdgcn_s_wait_asynccnt(N);
#else
  asm volatile("s_wait_asynccnt %0" ::"i"(N) : "memory");
#endif
  asm volatile("" ::: "memory");   // compiler barrier: LDS reads stay after wait
#endif
}

// Async-LDS writes are UNORDERED with the wave's own DS reads.
// Before the async engine may overwrite a
// buffer, drain our outstanding ds_loads from it.
__device__ __forceinline__ void drain_lds_reads() {
#if ASYNC_LDS
#if __has_builtin(__builtin_amdgcn_s_wait_dscnt)
  __builtin_amdgcn_s_wait_dscnt(0);
  asm volatile("" ::: "memory");
#else
  asm volatile("s_wait_dscnt 0" ::: "memory");
#endif
#endif
}

// Issue one stage: 4 x b128 async loads per thread, uniform issue count
// (addresses clamped so every thread always issues exactly 4 -> ASYNCcnt
// bookkeeping is exact with in-order completion).
__device__ __forceinline__ void issue_stage(const float* __restrict__ xrow,
                                            float* __restrict__ buf,
                                            int stage, int tid) {
  const int base = stage * CHUNK;
#pragma unroll
  for (int v = 0; v < 4; ++v) {
    int g = base + v * 1024 + tid * 4;
    if (g > C_COLS - 4) g = C_COLS - 4;  // clamp: duplicate load, ignored later
    float* dst = buf + tid * 16 + v * 4;
#if ASYNC_LDS
    __builtin_amdgcn_global_load_async_to_lds_b128(
        (gv4i_p)(const_cast<float*>(xrow) + g),
        (lv4i_p)dst,
        /*offset=*/0, /*cpol=*/STREAM_CPOL);
#else
    const float4 val = *(const float4*)(xrow + g);
    *(float4*)dst = val;
#endif
  }
}

__device__ __forceinline__ void online_update(float& m, float& s, float xv) {
  if (xv > m) {               // rare path: rescale running sum
    s = s * __expf(m - xv) + 1.0f;
    m = xv;
  } else {
    s += __expf(xv - m);
  }
}

__device__ __forceinline__ void consume_stage(const float* __restrict__ buf,
                                              int stage, int tid,
                                              float& m, float& s) {
  const int base = stage * CHUNK;
#pragma unroll
  for (int v = 0; v < 4; ++v) {
    const int g0 = base + v * 1024 + tid * 4;   // multiple of 4; C%4==0
    if (g0 < C_COLS) {
      const float4 d = *(const float4*)(buf + tid * 16 + v * 4);
      online_update(m, s, d.x);
      online_update(m, s, d.y);
      online_update(m, s, d.z);
      online_update(m, s, d.w);
    }
  }
}

__global__ void __launch_bounds__(TPB)
row_loss_kernel(const float* __restrict__ x, const int* __restrict__ y,
                float* __restrict__ ws) {
  __shared__ __align__(16) float buf[2][CHUNK];   // 32 KB double buffer
  __shared__ float red_m[TPB];
  __shared__ float red_s[TPB];
  __shared__ int   s_y[8];
  __shared__ float s_xg[8];

  const int tid = threadIdx.x;
  const int row = blockIdx.x;
  const float* xrow = x + (size_t)row * C_COLS;

  // Gather the <=8 label logits (t = y[0], relatives = y[1..7]).
  if (tid < 8) {
    const int v = y[row * 8 + tid];
    s_y[tid]  = v;
    s_xg[tid] = xrow[v < 0 ? 0 : v];
  }

  // ---- streamed online softmax over the row (async LDS pipeline) ----
  float m = -3.0e38f, s = 0.0f;
  issue_stage(xrow, buf[0], 0, tid);
  for (int st = 0; st < NSTAGES; ++st) {
    float* cur = buf[st & 1];
    float* nxt = buf[(st & 1) ^ 1];
    if (st + 1 < NSTAGES) {
      drain_lds_reads();                 // nxt buffer's old reads must be done
      issue_stage(xrow, nxt, st + 1, tid);
      wait_async<4>();   // 4 next-stage loads may remain; current stage done
    } else {
      wait_async<0>();
    }
    consume_stage(cur, st, tid, m, s);
  }

  // ---- combine (m,s) across the 256 threads ----
  red_m[tid] = m;
  red_s[tid] = s;
  __syncthreads();
  for (int off = TPB / 2; off > 0; off >>= 1) {
    if (tid < off) {
      const float m2 = red_m[tid + off], s2 = red_s[tid + off];
      const float M = fmaxf(m, m2);
      s = s * __expf(m - M) + s2 * __expf(m2 - M);
      m = M;
      red_m[tid] = m;
      red_s[tid] = s;
    }
    __syncthreads();
  }

  if (tid == 0) {
    const float M = m, S = s;
    const float logZ = M + __logf(S);
    const float xt   = s_xg[0];

    // Deduplicated excluded mass: distinct valid indices among {t, r1..r7}.
    float excl = 0.0f;
    for (int j = 0; j < 8; ++j) {
      const int idx = s_y[j];
      if (j > 0 && idx < 0) continue;
      bool dup = false;
      for (int k = 0; k < j; ++k) {
        const int i2 = s_y[k];
        if ((k == 0 || i2 >= 0) && i2 == idx) dup = true;
      }
      if (!dup) excl += __expf(s_xg[j] - M);
    }

    float minr = BIG_;
    bool  any  = false;
    for (int j = 1; j < 8; ++j) {
      if (s_y[j] >= 0) { minr = fminf(minr, s_xg[j]); any = true; }
    }

    const float maskedS = fmaxf(S - excl, 1e-37f);
    const float lse_mask = M + __logf(maskedS);

    float l2 = 0.0f, flag = 0.0f;
    if (any) {
      const float a = minr, b = lse_mask;
      const float mm = fmaxf(a, b);
      const float lse = mm + __logf(__expf(a - mm) + __expf(b - mm));
      l2 = lse - a;
      flag = 1.0f;
    }

    ws[row]              = logZ - xt;  // loss1 contribution
    ws[B_ROWS + row]     = l2;         // loss2 contribution
    ws[2 * B_ROWS + row] = flag;       // sample count contribution
  }
}

// Deterministic single-block tree reduction -> scalar.
__global__ void __launch_bounds__(TPB)
finalize_kernel(const float* __restrict__ ws, float* __restrict__ out) {
  __shared__ float sh0[TPB], sh1[TPB], sh2[TPB];
  const int tid = threadIdx.x;
  float s1 = 0.0f, s2 = 0.0f, s3 = 0.0f;
  for (int i = tid; i < B_ROWS; i += TPB) {
    s1 += ws[i];
    s2 += ws[B_ROWS + i];
    s3 += ws[2 * B_ROWS + i];
  }
  sh0[tid] = s1; sh1[tid] = s2; sh2[tid] = s3;
  __syncthreads();
  for (int off = TPB / 2; off > 0; off >>= 1) {
    if (tid < off) {
      sh0[tid] += sh0[tid + off];
      sh1[tid] += sh1[tid + off];
      sh2[tid] += sh2[tid + off];
    }
    __syncthreads();
  }
  if (tid == 0) {
    const float loss1 = sh0[0] / (float)B_ROWS;
    const float loss2 = sh1[0] / fmaxf(sh2[0], 1.0f);
    out[0] = loss1 + GAMMA_ * loss2;
  }
}

extern "C" void kernel_launch(void* const* d_in, const int* in_sizes, int n_in,
                              void* d_out, int out_size, void* d_ws, size_t ws_size,
                              hipStream_t stream) {
  const float* x = (const float*)d_in[0];   // (2048, 32000) f32
  const int*   y = (const int*)d_in[1];     // (2048, 8) i32
  float* ws  = (float*)d_ws;                // 3*2048 floats, rewritten every call
  float* out = (float*)d_out;               // scalar f32

  row_loss_kernel<<<B_ROWS, TPB, 0, stream>>>(x, y, ws);
  finalize_kernel<<<1, TPB, 0, stream>>>(ws, out);
}